// RelGraphConvLayer_12275016532442
// MI455X (gfx1250) — compile-verified
//
#include <hip/hip_runtime.h>

typedef __attribute__((ext_vector_type(2))) float v2f;
typedef __attribute__((ext_vector_type(8))) float v8f;

#define D_FEAT   128
#define LDSTRIDE 132                     // pad 128->132 floats: conflict-free dual-row LDS reads
#define WTILE    (D_FEAT * LDSTRIDE)     // one transposed weight matrix in LDS (floats)

// ---------------------------------------------------------------- zero scratch
__global__ void zero4_kernel(float4* __restrict__ p, long n4) {
    long i = (long)blockIdx.x * blockDim.x + threadIdx.x;
    long stride = (long)gridDim.x * blockDim.x;
    float4 z; z.x = z.y = z.z = z.w = 0.0f;
    for (; i < n4; i += stride) p[i] = z;
}

// ------------------------------------------------- edge scatter: wave per edge
__global__ void scatter_kernel(const float* __restrict__ xsrc,
                               const int*   __restrict__ src,
                               const int*   __restrict__ dst,
                               float*       __restrict__ msg,
                               float*       __restrict__ cnt,
                               int n_edges) {
    int gid  = blockIdx.x * blockDim.x + threadIdx.x;
    int e    = gid >> 5;                 // wave-uniform
    if (e >= n_edges) return;            // full-wave exit
    int lane = gid & 31;
    int s = src[e];
    int d = dst[e];
    float4 v = ((const float4*)(xsrc + (size_t)s * D_FEAT))[lane];
    float* m = msg + (size_t)d * D_FEAT + lane * 4;
    unsafeAtomicAdd(m + 0, v.x);         // global_atomic_add_f32
    unsafeAtomicAdd(m + 1, v.y);
    unsafeAtomicAdd(m + 2, v.z);
    unsafeAtomicAdd(m + 3, v.w);
    if (lane == 0) unsafeAtomicAdd(cnt + d, 1.0f);
}

// ------------------------------------------- counts -> reciprocal of max(c,1)
__global__ void invcnt_kernel(float* cnt, int n) {
    int i = blockIdx.x * blockDim.x + threadIdx.x;
    if (i < n) cnt[i] = 1.0f / fmaxf(cnt[i], 1.0f);
}

// --------------------------------------------------------------- fused SAGE GEMM
// out[r,:] = relu( sum_rel( (msg_rel[r,:] * inv_rel[r]) @ Wl_rel + bl_rel + x[r,:] @ Wr_rel ) )
// One wave computes a 16x128 output row-block with V_WMMA_F32_16X16X4_F32.
template<bool TWO>
__global__ void sage_gemm_kernel(const float* __restrict__ msg1,
                                 const float* __restrict__ inv1,
                                 const float* __restrict__ msg2,
                                 const float* __restrict__ inv2,
                                 const float* __restrict__ xdst,
                                 const float* __restrict__ Wl1,
                                 const float* __restrict__ Wr1,
                                 const float* __restrict__ Wl2,
                                 const float* __restrict__ Wr2,
                                 const float* __restrict__ bl1,
                                 const float* __restrict__ bl2,
                                 float*       __restrict__ out,
                                 int n_rows) {
    extern __shared__ float smem[];      // transposed weights: S[c*LDSTRIDE + k] = W[k,c]
    const int tid = threadIdx.x;
    {
        const float* Ws[4] = { Wl1, Wr1, Wl2, Wr2 };
        const int nw = TWO ? 4 : 2;
        for (int w = 0; w < nw; ++w) {
            float* S = smem + w * WTILE;
            const float* W = Ws[w];
            for (int idx = tid; idx < D_FEAT * D_FEAT; idx += blockDim.x) {
                int k = idx >> 7;        // coalesced global read, transposed LDS write
                int c = idx & 127;
                S[c * LDSTRIDE + k] = W[idx];
            }
        }
    }
    __syncthreads();

    int wave = blockIdx.x * (blockDim.x >> 5) + (tid >> 5);
    int row0 = wave * 16;                // wave-uniform
    if (row0 >= n_rows) return;          // full-wave exit -> EXEC all ones for WMMA

    const int lane = tid & 31;
    const int half = lane >> 4;          // lane split per ISA f32 WMMA layout
    const int lm   = lane & 15;
    const int rowA = row0 + lm;          // A-fragment row for this lane (both halves)

    v8f acc[8] = {};                     // 8 column tiles -> 8 independent WMMA chains

    auto mac = [&](const float* __restrict__ arow, float scale,
                   const float* __restrict__ S0, const float* __restrict__ S1) {
        const float2* ap = (const float2*)arow;
        for (int kk = 0; kk < 32; ++kk) {
            // A 16x4 f32: low lanes K=k0,k0+1 ; high lanes K=k0+2,k0+3
            float2 av = ap[2 * kk + half];
            v2f a; a.x = av.x * scale; a.y = av.y * scale;
            const int kbase = 4 * kk + 2 * half;
            #pragma unroll
            for (int nt = 0; nt < 8; ++nt) {
                const int col = nt * 16 + lm;
                // B 4x16 f32 from transposed LDS: b = (W[kbase,col], W[kbase+1,col])
                float2 bv = *(const float2*)(S0 + col * LDSTRIDE + kbase);  // ds_load_b64
                v2f b; b.x = bv.x; b.y = bv.y;
                acc[nt] = __builtin_amdgcn_wmma_f32_16x16x4_f32(
                    false, a, false, b, (short)0, acc[nt], false, false);
                if (S1) {                // reuse A fragment for second relation's Wr
                    float2 bw = *(const float2*)(S1 + col * LDSTRIDE + kbase);
                    v2f b2; b2.x = bw.x; b2.y = bw.y;
                    acc[nt] = __builtin_amdgcn_wmma_f32_16x16x4_f32(
                        false, a, false, b2, (short)0, acc[nt], false, false);
                }
            }
        }
    };

    mac(msg1 + (size_t)rowA * D_FEAT, inv1[rowA], smem + 0 * WTILE, nullptr);
    if (TWO)
        mac(msg2 + (size_t)rowA * D_FEAT, inv2[rowA], smem + 2 * WTILE, nullptr);
    mac(xdst + (size_t)rowA * D_FEAT, 1.0f, smem + 1 * WTILE,
        TWO ? smem + 3 * WTILE : nullptr);

    // epilogue: bias (summed over relations) + relu, D layout: VGPR i -> M = i + 8*half
    #pragma unroll
    for (int nt = 0; nt < 8; ++nt) {
        const int col = nt * 16 + lm;
        float bias = bl1[col] + (TWO ? bl2[col] : 0.0f);
        #pragma unroll
        for (int i = 0; i < 8; ++i) {
            float v = acc[nt][i] + bias;
            v = v > 0.0f ? v : 0.0f;
            out[(size_t)(row0 + half * 8 + i) * D_FEAT + col] = v;
        }
    }
}

// --------------------------------------------------------------------- launch
extern "C" void kernel_launch(void* const* d_in, const int* in_sizes, int n_in,
                              void* d_out, int out_size, void* d_ws, size_t ws_size,
                              hipStream_t stream) {
    const float* x_paper   = (const float*)d_in[0];
    const float* x_author  = (const float*)d_in[1];
    const float* x_inst    = (const float*)d_in[2];
    const int* cites_src   = (const int*)d_in[3];
    const int* cites_dst   = (const int*)d_in[4];
    const int* writes_src  = (const int*)d_in[5];
    const int* writes_dst  = (const int*)d_in[6];
    const int* rev_src     = (const int*)d_in[7];
    const int* rev_dst     = (const int*)d_in[8];
    const int* aff_src     = (const int*)d_in[9];
    const int* aff_dst     = (const int*)d_in[10];
    const float* Wl_cites  = (const float*)d_in[11];
    const float* bl_cites  = (const float*)d_in[12];
    const float* Wr_cites  = (const float*)d_in[13];
    const float* Wl_writes = (const float*)d_in[14];
    const float* bl_writes = (const float*)d_in[15];
    const float* Wr_writes = (const float*)d_in[16];
    const float* Wl_rev    = (const float*)d_in[17];
    const float* bl_rev    = (const float*)d_in[18];
    const float* Wr_rev    = (const float*)d_in[19];
    const float* Wl_aff    = (const float*)d_in[20];
    const float* bl_aff    = (const float*)d_in[21];
    const float* Wr_aff    = (const float*)d_in[22];

    const int NP = in_sizes[0] / D_FEAT;
    const int NA = in_sizes[1] / D_FEAT;
    const int NI = in_sizes[2] / D_FEAT;
    const int EC = in_sizes[3];
    const int EW = in_sizes[5];
    const int ER = in_sizes[7];
    const int EA = in_sizes[9];

    // workspace layout
    float* ws   = (float*)d_ws;
    float* msgC = ws; ws += (size_t)NP * D_FEAT;
    float* msgW = ws; ws += (size_t)NP * D_FEAT;
    float* msgR = ws; ws += (size_t)NA * D_FEAT;
    float* msgA = ws; ws += (size_t)NI * D_FEAT;
    float* cntC = ws; ws += NP;
    float* cntW = ws; ws += NP;
    float* cntR = ws; ws += NA;
    float* cntA = ws; ws += NI;
    const size_t total_f = (size_t)(ws - (float*)d_ws);

    float* h_paper  = (float*)d_out;
    float* h_author = h_paper + (size_t)NP * D_FEAT;
    float* h_inst   = h_author + (size_t)NA * D_FEAT;

    const int B = 256;

    // 1) zero accumulation scratch (required every launch: graph replays)
    long n4 = (long)(total_f / 4);
    long zblocks = (n4 + B - 1) / B;
    zero4_kernel<<<(unsigned)zblocks, B, 0, stream>>>((float4*)d_ws, n4);

    // 2) edge scatter (wave per edge)
    auto sblocks = [&](int e) { return (unsigned)(((long)e * 32 + B - 1) / B); };
    scatter_kernel<<<sblocks(EC), B, 0, stream>>>(x_paper,  cites_src,  cites_dst,  msgC, cntC, EC);
    scatter_kernel<<<sblocks(EW), B, 0, stream>>>(x_author, writes_src, writes_dst, msgW, cntW, EW);
    scatter_kernel<<<sblocks(ER), B, 0, stream>>>(x_paper,  rev_src,    rev_dst,    msgR, cntR, ER);
    scatter_kernel<<<sblocks(EA), B, 0, stream>>>(x_author, aff_src,    aff_dst,    msgA, cntA, EA);

    // 3) counts -> reciprocals
    invcnt_kernel<<<(NP + B - 1) / B, B, 0, stream>>>(cntC, NP);
    invcnt_kernel<<<(NP + B - 1) / B, B, 0, stream>>>(cntW, NP);
    invcnt_kernel<<<(NA + B - 1) / B, B, 0, stream>>>(cntR, NA);
    invcnt_kernel<<<(NI + B - 1) / B, B, 0, stream>>>(cntA, NI);

    // 4) fused WMMA GEMM + bias + relu per dst node type
    const size_t smem2 = (size_t)2 * WTILE * sizeof(float);   // 135168 B
    const size_t smem4 = (size_t)4 * WTILE * sizeof(float);   // 270336 B (<320KB WGP LDS)
    hipFuncSetAttribute(reinterpret_cast<const void*>(sage_gemm_kernel<true>),
                        hipFuncAttributeMaxDynamicSharedMemorySize, (int)smem4);
    hipFuncSetAttribute(reinterpret_cast<const void*>(sage_gemm_kernel<false>),
                        hipFuncAttributeMaxDynamicSharedMemorySize, (int)smem2);

    auto gblocks = [&](int n) { return (unsigned)(((n + 15) / 16 + 7) / 8); };

    // paper: cites (paper->paper) + writes (author->paper)
    sage_gemm_kernel<true><<<gblocks(NP), B, smem4, stream>>>(
        msgC, cntC, msgW, cntW, x_paper,
        Wl_cites, Wr_cites, Wl_writes, Wr_writes, bl_cites, bl_writes,
        h_paper, NP);
    // author: rev (paper->author)
    sage_gemm_kernel<false><<<gblocks(NA), B, smem2, stream>>>(
        msgR, cntR, nullptr, nullptr, x_author,
        Wl_rev, Wr_rev, nullptr, nullptr, bl_rev, nullptr,
        h_author, NA);
    // inst: aff (author->inst)
    sage_gemm_kernel<false><<<gblocks(NI), B, smem2, stream>>>(
        msgA, cntA, nullptr, nullptr, x_inst,
        Wl_aff, Wr_aff, nullptr, nullptr, bl_aff, nullptr,
        h_inst, NI);
}